// SoftPerspectiveUnshader_10007273800428
// MI455X (gfx1250) — compile-verified
//
#include <hip/hip_runtime.h>
#include <hip/hip_bf16.h>
#include <stdint.h>

#define NB 4
#define HH 512
#define WW 512
#define KK 4
#define CC 3
#define PIX (NB * HH * WW)        // 1048576 pixels
#define BLK 256
#define NBLK (PIX / BLK)          // 4096
#define ACC_FLOATS (PIX * 4)      // [N,H,W,4] rgb+wsum interleaved, 16 MB

#define SIGMA_INV 1.0e4f
#define GAMMA_INV 1.0e4f
#define ZSCALE (1.0f / 99.0f)     // 1/(ZFAR-ZNEAR)
#define ZFAR_C 100.0f
#define DELTA_C 1.000001000001e-10f   // exp(1e-6)*1e-10

typedef unsigned int u32x4 __attribute__((ext_vector_type(4)));
typedef unsigned int u32x8 __attribute__((ext_vector_type(8)));

// low 32 bits of a flat LDS pointer == byte offset within the workgroup's LDS
__device__ __forceinline__ unsigned lds_off_of(const void* p) {
  return (unsigned)(unsigned long long)(uintptr_t)p;
}

// ---- CDNA5 async global->LDS copy (ASYNCcnt path, gfx1250) --------------
__device__ __forceinline__ void async_copy16(void* lds_dst, const void* gsrc) {
  unsigned lds_off = lds_off_of(lds_dst);
  asm volatile("global_load_async_to_lds_b128 %0, %1, off"
               :: "v"(lds_off), "v"(gsrc)
               : "memory");
}
__device__ __forceinline__ void wait_async0() {
  asm volatile("s_wait_asynccnt 0x0" ::: "memory");
}
__device__ __forceinline__ void wait_tensor0() {
#if __has_builtin(__builtin_amdgcn_s_wait_tensorcnt)
  __builtin_amdgcn_s_wait_tensorcnt(0);
#else
  asm volatile("s_wait_tensorcnt 0x0" ::: "memory");
#endif
}

// ---- CDNA5 Tensor Data Mover: 1-D contiguous tile, global -> LDS --------
// D# group0: [1:0]count=1 | [63:32]lds_addr | [120:64]global_addr | [127:126]type=2
// D# group1: [17:16]data_size=3(8B) | [79:48]tensor_dim0=nelem | [111:80]tensor_dim1=1
//            | [127:112]tile_dim0=nelem | [143:128]tile_dim1=1 | [159:144]tile_dim2=0
//            | [207:160]tensor_dim0_stride=nelem
__device__ __forceinline__ void tdm_load_lds_1d(unsigned lds_off, const void* gsrc,
                                                unsigned nelem8 /* 8-byte elems */) {
  const unsigned long long ga = (unsigned long long)(uintptr_t)gsrc;
  u32x4 g0;
  g0[0] = 1u;                                   // count=1, user descriptor
  g0[1] = lds_off;                              // lds_addr
  g0[2] = (unsigned)ga;                         // global_addr[31:0]
  g0[3] = ((unsigned)(ga >> 32) & 0x01FFFFFFu)  // global_addr[56:32]
          | (2u << 30);                         // type=2 ("image")
  u32x8 g1;
  g1[0] = 3u << 16;                             // data_size=8B, mask=0, no pad/iter
  g1[1] = (nelem8 & 0xFFFFu) << 16;             // tensor_dim0[15:0]
  g1[2] = (nelem8 >> 16) | (1u << 16);          // tensor_dim0[31:16] | tensor_dim1=1
  g1[3] = (nelem8 & 0xFFFFu) << 16;             // tensor_dim1[31:16]=0 | tile_dim0
  g1[4] = 1u;                                   // tile_dim1=1 (explicit), tile_dim2=0
  g1[5] = nelem8;                               // tensor_dim0_stride[31:0]
  g1[6] = 0u;                                   // stride0[47:32]=0, stride1[15:0]=0
  g1[7] = 0u;                                   // stride1 hi=0
  asm volatile("tensor_load_to_lds %0, %1" :: "s"(g0), "s"(g1) : "memory");
}

__device__ __forceinline__ void atom_add_f32(float* p, float v) {
  unsafeAtomicAdd(p, v);   // global_atomic_add_f32 (no return)
}

// ---- kernel 1: zero accumulator -----------------------------------------
__global__ void zero_acc_kernel(float4* __restrict__ acc) {
  size_t i = (size_t)blockIdx.x * blockDim.x + threadIdx.x;
  acc[i] = make_float4(0.f, 0.f, 0.f, 0.f);
}

// ---- kernel 2: softmax blend + UV + bilinear scatter-add ----------------
struct Smem {
  int   p2f [BLK * KK];        // 4096 B
  float bary[BLK * KK * 3];    // 12288 B
  float dist[BLK * KK];        // 4096 B
  float zbuf[BLK * KK];        // 4096 B
  float img [BLK * CC];        // 3072 B
};

__global__ __launch_bounds__(BLK) void splat_kernel(
    const int*   __restrict__ p2f,
    const float* __restrict__ bary,
    const float* __restrict__ dists,
    const float* __restrict__ zbuf,
    const float* __restrict__ images,
    const float* __restrict__ fuv,
    float*       __restrict__ acc) {
  __shared__ Smem sm;
  const int tid = threadIdx.x;
  const size_t pbase = (size_t)blockIdx.x * BLK;

  // ---- stage per-pixel streamed inputs into LDS ----
  // Wave 0 issues TDM descriptors for the bulk (24.5 KB) of the staging;
  // images (12 B/pixel) go through per-lane async-LDS b128 copies.
  if (tid < 32) {
    tdm_load_lds_1d(lds_off_of(sm.p2f),  (const char*)p2f   + pbase * 16, (BLK * KK * 4) / 8);
    tdm_load_lds_1d(lds_off_of(sm.bary), (const char*)bary  + pbase * 48, (BLK * 12 * 4) / 8);
    tdm_load_lds_1d(lds_off_of(sm.dist), (const char*)dists + pbase * 16, (BLK * KK * 4) / 8);
    tdm_load_lds_1d(lds_off_of(sm.zbuf), (const char*)zbuf  + pbase * 16, (BLK * KK * 4) / 8);
  }
  if (tid < (BLK * CC * 4) / 16)   // 192 chunks of 16 B
    async_copy16((char*)sm.img + tid * 16,
                 (const char*)images + pbase * 12 + tid * 16);
  wait_tensor0();   // all waves: no-op unless this wave issued TDM ops
  wait_async0();
  __syncthreads();

  const int p  = (int)pbase + tid;
  const int n  = p >> 18;            // H*W = 2^18
  const int hw = p & ((1 << 18) - 1);
  const int y  = hw >> 9;            // W = 2^9
  const int x  = hw & 511;

  // fragment faces + prefetch the 4 random UV rows (L2-resident 3 MB table)
  int fk[KK];
#pragma unroll
  for (int k = 0; k < KK; ++k) {
    fk[k] = sm.p2f[tid * KK + k];
    if (fk[k] >= 0)
      __builtin_prefetch(fuv + (size_t)fk[k] * 6, 0, 0);  // global_prefetch_b8
  }

  // ---- softmax blend weights over K ----
  float wnum[KK], zi[KK];
#pragma unroll
  for (int k = 0; k < KK; ++k) {
    const bool  m = fk[k] >= 0;
    const float d = sm.dist[tid * KK + k];
    const float z = sm.zbuf[tid * KK + k];
    wnum[k] = m ? 1.0f / (1.0f + expf(d * SIGMA_INV)) : 0.0f;   // prob
    zi[k]   = m ? (ZFAR_C - z) * ZSCALE : 0.0f;
  }
  const float zmax = fmaxf(fmaxf(zi[0], zi[1]), fmaxf(zi[2], zi[3]));
  float denom = DELTA_C;
#pragma unroll
  for (int k = 0; k < KK; ++k) {
    wnum[k] *= expf((zi[k] - zmax) * GAMMA_INV);  // underflows to 0 for losers
    denom += wnum[k];
  }
  const float rcp = 1.0f / denom;

  const float c0 = sm.img[tid * CC + 0];
  const float c1 = sm.img[tid * CC + 1];
  const float c2 = sm.img[tid * CC + 2];
  float* accn = acc + ((size_t)n << 20);   // n * H*W*4

  // ---- per-fragment UV target + bilinear atomic scatter ----
#pragma unroll
  for (int k = 0; k < KK; ++k) {
    const float wk = wnum[k] * rcp;
    if (fk[k] < 0 || wk == 0.0f) continue;   // masked / exp-underflow: adds 0
    const float* uv = fuv + (size_t)fk[k] * 6;
    const float b0 = sm.bary[tid * 12 + k * 3 + 0];
    const float b1 = sm.bary[tid * 12 + k * 3 + 1];
    const float b2 = sm.bary[tid * 12 + k * 3 + 2];
    const float u = b0 * uv[0] + b1 * uv[2] + b2 * uv[4];
    const float v = b0 * uv[1] + b1 * uv[3] + b2 * uv[5];
    // fx = (u*W - x*W/(W-1)) + x  (matches reference association)
    const float fx = (u * 512.0f - (float)x * (512.0f / 511.0f)) + (float)x;
    const float fy = (v * 512.0f - (float)y * (512.0f / 511.0f)) + (float)y;
    const float x0f = floorf(fx), y0f = floorf(fy);
    const float tx = fx - x0f,    ty = fy - y0f;
    const int   x0 = (int)x0f,    y0 = (int)y0f;
    const float wtx[2] = {1.0f - tx, tx};
    const float wty[2] = {1.0f - ty, ty};
#pragma unroll
    for (int dy = 0; dy < 2; ++dy) {
      const int yi = y0 + dy;
      if ((unsigned)yi >= 512u) continue;
#pragma unroll
      for (int dx = 0; dx < 2; ++dx) {
        const int xi = x0 + dx;
        if ((unsigned)xi >= 512u) continue;
        const float g = wk * wtx[dx] * wty[dy];
        float* a = accn + ((((size_t)yi << 9) + xi) << 2);
        atom_add_f32(a + 0, c0 * g);
        atom_add_f32(a + 1, c1 * g);
        atom_add_f32(a + 2, c2 * g);
        atom_add_f32(a + 3, g);
      }
    }
  }
}

// ---- kernel 3: normalize + H-flip + write (textures, weights) -----------
__global__ __launch_bounds__(BLK) void finalize_kernel(
    const float* __restrict__ acc, float* __restrict__ out) {
  const int p  = blockIdx.x * BLK + threadIdx.x;
  const int n  = p >> 18;
  const int hw = p & ((1 << 18) - 1);
  const int h  = hw >> 9;
  const int w  = hw & 511;
  const float4 a = *(const float4*)(acc + ((size_t)n << 20) +
                                    ((((size_t)(511 - h) << 9) + w) << 2));
  const float ws  = a.w;
  const float inv = 1.0f / fmaxf(ws, 1e-8f);
  float* tex = out + (size_t)p * 3;
  tex[0] = a.x * inv;
  tex[1] = a.y * inv;
  tex[2] = a.z * inv;
  float* wo = out + (size_t)PIX * 3 + (size_t)p * 3;
  wo[0] = ws; wo[1] = ws; wo[2] = ws;
}

extern "C" void kernel_launch(void* const* d_in, const int* in_sizes, int n_in,
                              void* d_out, int out_size, void* d_ws, size_t ws_size,
                              hipStream_t stream) {
  const int*   p2f    = (const int*)  d_in[0];
  const float* bary   = (const float*)d_in[1];
  const float* dists  = (const float*)d_in[2];
  const float* zbuf   = (const float*)d_in[3];
  const float* images = (const float*)d_in[4];
  const float* fuv    = (const float*)d_in[5];
  float* acc = (float*)d_ws;                 // needs 16 MB scratch
  float* out = (float*)d_out;

  zero_acc_kernel<<<ACC_FLOATS / 4 / BLK, BLK, 0, stream>>>((float4*)acc);
  splat_kernel<<<NBLK, BLK, 0, stream>>>(p2f, bary, dists, zbuf, images, fuv, acc);
  finalize_kernel<<<NBLK, BLK, 0, stream>>>(acc, out);
}